// bone2joint_7954279432434
// MI455X (gfx1250) — compile-verified
//
#include <hip/hip_runtime.h>
#include <stdint.h>

typedef __attribute__((ext_vector_type(2))) float v2f;
typedef __attribute__((ext_vector_type(8))) float v8f;

// Path masks: bit k of kPathMask[j] == 1 iff bone k lies on path center(1) -> joint j.
// Rows 25..31 are zero padding for the 32-row WMMA M-tiling.
__device__ const unsigned kPathMask[32] = {
    0x00000001u, // 0: {0}
    0x00000000u, // 1: center
    0x00100004u, // 2: {2,20}
    0x0010000Cu, // 3: {3,2,20}
    0x00100010u, // 4: {4,20}
    0x00100030u, // 5
    0x00100070u, // 6
    0x001000F0u, // 7
    0x00100100u, // 8
    0x00100300u, // 9
    0x00100700u, // 10
    0x00100F00u, // 11
    0x00001001u, // 12: {12,0}
    0x00003001u, // 13
    0x00007001u, // 14
    0x0000F001u, // 15
    0x00010001u, // 16
    0x00030001u, // 17
    0x00070001u, // 18
    0x000F0001u, // 19
    0x00100000u, // 20
    0x003000F0u, // 21
    0x005000F0u, // 22
    0x00900F00u, // 23
    0x01100F00u, // 24
    0, 0, 0, 0, 0, 0, 0
};

#define WAVES_PER_BLOCK 4
#define SLICES_PER_WAVE 16
#define FLOATS_PER_WAVE (SLICES_PER_WAVE * 75)  // 1200
#define VEC4_PER_WAVE   (FLOATS_PER_WAVE / 4)   // 300
#define LDS_BONE_STRIDE 1232                    // padded so OOB K reads stay in-slab

__global__ __launch_bounds__(WAVES_PER_BLOCK * 32)
void bone2joint_wmma_kernel(const float* __restrict__ bone,
                            const float* __restrict__ center,
                            float* __restrict__ joint,
                            unsigned nSlices) {
    __shared__ float sBone[WAVES_PER_BLOCK][LDS_BONE_STRIDE];
    __shared__ float sOut [WAVES_PER_BLOCK][FLOATS_PER_WAVE];

    const unsigned tid  = threadIdx.x;
    const unsigned lane = tid & 31u;
    const unsigned w    = tid >> 5;
    const unsigned hi   = lane >> 4;   // lane half selects K+2 / row+8
    const unsigned l16  = lane & 15u;

    const unsigned sliceBase = (blockIdx.x * WAVES_PER_BLOCK + w) * SLICES_PER_WAVE;
    if (sliceBase >= nSlices) return;

    const float* gBone = bone  + (size_t)sliceBase * 75u;
    float*       gOut  = joint + (size_t)sliceBase * 75u;

    if (sliceBase + SLICES_PER_WAVE > nSlices) {
        // Scalar tail path (wave-uniform branch; not hit for 307200 slices).
        const unsigned nElem = (nSlices - sliceBase) * 75u;
        for (unsigned e = lane; e < nElem; e += 32u) {
            const unsigned s = e / 75u, r = e - 75u * s;
            const unsigned j = r / 3u,  c = r - 3u * j;
            float acc = center[(size_t)(sliceBase + s) * 3u + c];
            unsigned m = kPathMask[j];
            while (m) {
                const int k = __ffs(m) - 1;
                m &= m - 1u;
                acc += gBone[s * 75u + (unsigned)k * 3u + c];
            }
            gOut[e] = acc;
        }
        return;
    }

    // ---- Async DMA: bone slab (300 x b128, 16B aligned) global -> LDS ----
    const uint32_t ldsBoneBase = (uint32_t)(uintptr_t)(void*)&sBone[w][0];
    for (unsigned i = lane; i < VEC4_PER_WAVE; i += 32u) {
        const uint32_t ldsOff = ldsBoneBase + i * 16u;
        const uint32_t gOff   = i * 16u;
        asm volatile("global_load_async_to_lds_b128 %0, %1, %2"
                     :: "v"(ldsOff), "v"(gOff), "s"(gBone) : "memory");
    }

    // ---- A-matrix fragments from path masks (constant, registers only) ----
    // A frag layout (16x4 f32): lanes 0-15 M=l16 {K,K+1}; lanes 16-31 {K+2,K+3}.
    const unsigned m0 = kPathMask[l16];        // M-tile 0: rows 0..15
    const unsigned m1 = kPathMask[16u + l16];  // M-tile 1: rows 16..31 (25+ zero)
    v2f a0[7], a1[7];
#pragma unroll
    for (int kt = 0; kt < 7; ++kt) {
        const unsigned kk = (unsigned)kt * 4u + hi * 2u;
        a0[kt].x = ((m0 >> kk) & 1u) ? 1.0f : 0.0f;
        a0[kt].y = ((m0 >> (kk + 1u)) & 1u) ? 1.0f : 0.0f;
        a1[kt].x = ((m1 >> kk) & 1u) ? 1.0f : 0.0f;
        a1[kt].y = ((m1 >> (kk + 1u)) & 1u) ? 1.0f : 0.0f;
    }

    asm volatile("s_wait_asynccnt 0" ::: "memory");

    // ---- 3 column panels of 16 columns; column = (slice, c) pair ----
#pragma unroll
    for (int p = 0; p < 3; ++p) {
        const unsigned col = (unsigned)p * 16u + l16;
        const unsigned s = col / 3u, c = col - 3u * s;
        const unsigned colbase = s * 75u + c;

        v8f acc0 = {};
        v8f acc1 = {};
#pragma unroll
        for (int kt = 0; kt < 7; ++kt) {
            const unsigned kk = (unsigned)kt * 4u + hi * 2u;
            // B frag (4x16): VGPR v holds rows {v, v+2} striped over lane halves.
            v2f b;
            float bx = sBone[w][colbase + kk * 3u];           // padded slab: safe
            float by = sBone[w][colbase + kk * 3u + 3u];
            b.x = (kk < 25u) ? bx : 0.0f;
            b.y = (kk + 1u < 25u) ? by : 0.0f;
            acc0 = __builtin_amdgcn_wmma_f32_16x16x4_f32(
                false, a0[kt], false, b, (short)0, acc0, false, false);
            acc1 = __builtin_amdgcn_wmma_f32_16x16x4_f32(
                false, a1[kt], false, b, (short)0, acc1, false, false);
        }

        const float cv = center[(size_t)sliceBase * 3u + col];
        // D layout: VGPR r -> rows {r, r+8} over lane halves (+16 for tile 1).
#pragma unroll
        for (int r = 0; r < 8; ++r) {
            const unsigned row0 = (unsigned)r + hi * 8u;          // 0..15
            sOut[w][colbase + row0 * 3u] = acc0[r] + cv;
            const unsigned row1 = 16u + (unsigned)r + hi * 8u;    // 16..31
            if (row1 < 25u)
                sOut[w][colbase + row1 * 3u] = acc1[r] + cv;
        }
    }

    // ---- Async DMA: joint slab LDS -> global (coalesced b128 stream) ----
    asm volatile("s_wait_dscnt 0" ::: "memory");
    const uint32_t ldsOutBase = (uint32_t)(uintptr_t)(void*)&sOut[w][0];
    for (unsigned i = lane; i < VEC4_PER_WAVE; i += 32u) {
        const uint32_t ldsOff = ldsOutBase + i * 16u;
        const uint32_t gOff   = i * 16u;
        asm volatile("global_store_async_from_lds_b128 %0, %1, %2"
                     :: "v"(gOff), "v"(ldsOff), "s"(gOut) : "memory");
    }
    asm volatile("s_wait_asynccnt 0" ::: "memory");
}

extern "C" void kernel_launch(void* const* d_in, const int* in_sizes, int n_in,
                              void* d_out, int out_size, void* d_ws, size_t ws_size,
                              hipStream_t stream) {
    const float* bone   = (const float*)d_in[0];
    const float* center = (const float*)d_in[1];
    float* joint = (float*)d_out;

    const unsigned nSlices = (unsigned)(in_sizes[1] / 3);  // B*T from center[B,T,3]
    const unsigned slicesPerBlock = WAVES_PER_BLOCK * SLICES_PER_WAVE;  // 64
    const unsigned blocks = (nSlices + slicesPerBlock - 1u) / slicesPerBlock;

    bone2joint_wmma_kernel<<<dim3(blocks), dim3(WAVES_PER_BLOCK * 32), 0, stream>>>(
        bone, center, joint, nSlices);
}